// Similarity_9320079032492
// MI455X (gfx1250) — compile-verified
//
#include <hip/hip_runtime.h>

typedef float v2f __attribute__((ext_vector_type(2)));
typedef float v8f __attribute__((ext_vector_type(8)));

constexpr int B = 32, T = 1024, J = 256, D = 512;
constexpr int TM = 64, TN = 64, KC = 32;
constexpr int LDA = KC + 4; // 36 floats: conflict-free fragment reads, 16B-aligned rows

// ---------------------------------------------------------------------------
// CDNA5 async global->LDS copy (ASYNCcnt-tracked, no VGPR staging).
// VDST = VGPR with 32-bit LDS byte address, VADDR = 64-bit global address.
// ---------------------------------------------------------------------------
__device__ __forceinline__ void async_copy_b128(const float* gsrc, unsigned lds_addr) {
    asm volatile("global_load_async_to_lds_b128 %0, %1, off"
                 :: "v"(lds_addr), "v"(gsrc) : "memory");
}
__device__ __forceinline__ void wait_async() {
    asm volatile("s_wait_asynccnt 0" ::: "memory");
}
__device__ __forceinline__ unsigned lds_addr_of(const void* p) {
    return (unsigned)(uintptr_t)p;   // generic shared ptr: LDS offset lives in addr[31:0]
}
__device__ __forceinline__ float4 fold4(float4 uv, float4 mv, float4 hv) {
    uv.x = fmaf(uv.x, mv.x, hv.x);
    uv.y = fmaf(uv.y, mv.y, hv.y);
    uv.z = fmaf(uv.z, mv.z, hv.z);
    uv.w = fmaf(uv.w, mv.w, hv.w);
    return uv;
}

// ---------------------------------------------------------------------------
// Pass 1: term_u[b,j] = u[b,j,:] . w_u   (one wave32 per row, 8 rows / block)
// ---------------------------------------------------------------------------
__global__ __launch_bounds__(256) void row_dot_kernel(const float* __restrict__ x,
                                                      const float* __restrict__ w,
                                                      float* __restrict__ out) {
    const int wave = threadIdx.x >> 5;
    const int lane = threadIdx.x & 31;
    const int row  = blockIdx.x * 8 + wave;
    const float* xr = x + (size_t)row * D;
    float s = 0.0f;
#pragma unroll
    for (int i = 0; i < 4; ++i) {            // 512 = 32 lanes * 4 * float4
        const int k = lane * 4 + i * 128;
        const float4 xv = *(const float4*)(xr + k);
        const float4 wv = *(const float4*)(w + k);
        s += xv.x * wv.x + xv.y * wv.y + xv.z * wv.z + xv.w * wv.w;
    }
#pragma unroll
    for (int off = 16; off > 0; off >>= 1) s += __shfl_xor(s, off, 32);
    if (lane == 0) out[row] = s;
}

// ---------------------------------------------------------------------------
// Pass 2: fused batched GEMM via V_WMMA_F32_16X16X4_F32, double-buffered LDS,
// h tile moved with GLOBAL_LOAD_ASYNC_TO_LDS_B128.
//   out[b,t,j] = sum_k h[b,t,k] * (w_m[k]*u[b,j,k] + w_h[k]) + term_u[b,j]
// ---------------------------------------------------------------------------
__global__ __launch_bounds__(256) void similarity_wmma_kernel(
    const float* __restrict__ h, const float* __restrict__ u,
    const float* __restrict__ w, const float* __restrict__ tu,
    float* __restrict__ out) {

    __shared__ float a_lds[2][TM * LDA];   // h tile, raw          [m][k]
    __shared__ float b_lds[2][TN * LDA];   // u*w_m + w_h tile     [n][k]

    const int t0 = blockIdx.x * TM;
    const int j0 = blockIdx.y * TN;
    const int b  = blockIdx.z;

    const float* wh = w;          // w[0:D]
    const float* wm = w + 2 * D;  // w[2D:3D]

    const int tid  = threadIdx.x;
    const int lane = tid & 31;
    const int wv   = tid >> 5;
    const int r16  = lane & 15;
    const int hi   = lane >> 4;

    const int mi  = wv & 3;          // row block 0..3 (16 rows each)
    const int nj0 = (wv >> 2) * 2;   // first of two col blocks
    const int nj1 = nj0 + 1;

    v8f acc0 = {};
    v8f acc1 = {};

    // cooperative tile loader mapping: 256 threads, 64 rows x 32 cols, 8 floats each
    const int lrow = tid >> 2;        // 0..63
    const int lcol = (tid & 3) * 8;   // 0,8,16,24

    const float* hbase = h + ((size_t)b * T + (t0 + lrow)) * D + lcol;
    const float* ubase = u + ((size_t)b * J + (j0 + lrow)) * D + lcol;

    const unsigned a_dst0 = lds_addr_of(&a_lds[0][lrow * LDA + lcol]);
    const unsigned a_dst1 = lds_addr_of(&a_lds[1][lrow * LDA + lcol]);
    float* const b_st0 = &b_lds[0][lrow * LDA + lcol];
    float* const b_st1 = &b_lds[1][lrow * LDA + lcol];

    // per-wave fragment read bases (row = lane&15, k = k0 + 2*(lane>>4))
    const float* const a_rd0 = &a_lds[0][(mi  * 16 + r16) * LDA + 2 * hi];
    const float* const a_rd1 = &a_lds[1][(mi  * 16 + r16) * LDA + 2 * hi];
    const float* const b0rd0 = &b_lds[0][(nj0 * 16 + r16) * LDA + 2 * hi];
    const float* const b0rd1 = &b_lds[1][(nj0 * 16 + r16) * LDA + 2 * hi];
    const float* const b1rd0 = &b_lds[0][(nj1 * 16 + r16) * LDA + 2 * hi];
    const float* const b1rd1 = &b_lds[1][(nj1 * 16 + r16) * LDA + 2 * hi];

    // ---- prologue: fill buffer 0 for kt = 0 ----
    async_copy_b128(hbase,     a_dst0);
    async_copy_b128(hbase + 4, a_dst0 + 16);
    {
        float4 u0 = *(const float4*)(ubase);
        float4 u1 = *(const float4*)(ubase + 4);
        u0 = fold4(u0, *(const float4*)(wm + lcol),     *(const float4*)(wh + lcol));
        u1 = fold4(u1, *(const float4*)(wm + lcol + 4), *(const float4*)(wh + lcol + 4));
        *(float4*)(b_st0)     = u0;
        *(float4*)(b_st0 + 4) = u1;
    }
    wait_async();
    __syncthreads();

    for (int kt = 0; kt < D; kt += KC) {
        const int  cur  = (kt >> 5) & 1;
        const bool more = (kt + KC) < D;

        // issue next tile's fills: async h -> the other LDS buffer, u -> VGPRs
        float4 u0n, u1n, m0n, m1n, v0n, v1n;
        if (more) {
            const float* hn = hbase + kt + KC;
            async_copy_b128(hn,     cur ? a_dst0 : a_dst1);
            async_copy_b128(hn + 4, cur ? (a_dst0 + 16) : (a_dst1 + 16));
            const int kn = kt + KC + lcol;
            u0n = *(const float4*)(ubase + kt + KC);
            u1n = *(const float4*)(ubase + kt + KC + 4);
            m0n = *(const float4*)(wm + kn);
            m1n = *(const float4*)(wm + kn + 4);
            v0n = *(const float4*)(wh + kn);
            v1n = *(const float4*)(wh + kn + 4);
        }

        // compute on buffer `cur` (overlaps the async copy + u loads above)
        const float* ap  = cur ? a_rd1 : a_rd0;
        const float* bp0 = cur ? b0rd1 : b0rd0;
        const float* bp1 = cur ? b1rd1 : b1rd0;
#pragma unroll
        for (int k0 = 0; k0 < KC; k0 += 4) {
            const v2f af = *(const v2f*)(ap  + k0);
            const v2f b0 = *(const v2f*)(bp0 + k0);
            const v2f b1 = *(const v2f*)(bp1 + k0);
            acc0 = __builtin_amdgcn_wmma_f32_16x16x4_f32(false, af, false, b0,
                                                         (short)0, acc0, false, false);
            acc1 = __builtin_amdgcn_wmma_f32_16x16x4_f32(false, af, false, b1,
                                                         (short)0, acc1, false, false);
        }

        // fold + stage next u tile into the other LDS buffer
        if (more) {
            float* bs = cur ? b_st0 : b_st1;
            u0n = fold4(u0n, m0n, v0n);
            u1n = fold4(u1n, m1n, v1n);
            *(float4*)(bs)     = u0n;
            *(float4*)(bs + 4) = u1n;
        }

        wait_async();        // h tile for next iteration is in LDS
        __syncthreads();     // (also orders the ds_store'd u tile)
    }

    // epilogue: add term_u and store. C/D layout: VGPR r, lane l -> M=r+8*(l>>4), N=l&15
    const float* tub = tu + (size_t)b * J + j0;
    const float tu0 = tub[nj0 * 16 + r16];
    const float tu1 = tub[nj1 * 16 + r16];
    float* ob = out + ((size_t)b * T + t0) * J + j0;
#pragma unroll
    for (int r = 0; r < 8; ++r) {
        const int m = mi * 16 + hi * 8 + r;
        ob[(size_t)m * J + nj0 * 16 + r16] = acc0[r] + tu0;
        ob[(size_t)m * J + nj1 * 16 + r16] = acc1[r] + tu1;
    }
}

// ---------------------------------------------------------------------------
extern "C" void kernel_launch(void* const* d_in, const int* in_sizes, int n_in,
                              void* d_out, int out_size, void* d_ws, size_t ws_size,
                              hipStream_t stream) {
    const float* h = (const float*)d_in[0];
    const float* u = (const float*)d_in[1];
    const float* w = (const float*)d_in[2];
    float* out = (float*)d_out;
    float* tu  = (float*)d_ws;              // B*J floats of workspace

    // term_u[b,j] = u . w_u  (w_u = w[D:2D])
    row_dot_kernel<<<(B * J) / 8, 256, 0, stream>>>(u, w + D, tu);

    dim3 grid(T / TM, J / TN, B);
    similarity_wmma_kernel<<<grid, 256, 0, stream>>>(h, u, w, tu, out);
}